// GNN_91036126806390
// MI455X (gfx1250) — compile-verified
//
#include <hip/hip_runtime.h>
#include <hip/hip_bf16.h>

// ---------------- problem constants (match reference) ----------------
#define LNUM   3
#define EDIM   64
#define H2     128      // 2*EDIM
#define BGR    4
#define NNODE  20000
#define NEDGE  320000
#define INDIM  21
#define BN_EPS 1e-5f

typedef __attribute__((ext_vector_type(2))) float v2f;
typedef __attribute__((ext_vector_type(8))) float v8f;

// ---------------------------------------------------------------------
// Async (VGPR-bypassing) stage of weights into LDS. CDNA5 path:
// GLOBAL_LOAD_ASYNC_TO_LDS_B128, tracked by ASYNCcnt.
// LDS byte offset = low 32 bits of the generic pointer (high dword is the
// shared aperture per the ISA aperture check).
__device__ __forceinline__ void async_stage(const float* __restrict__ g,
                                            float* lds, int nfloats) {
    int nvec = nfloats >> 2;                       // 16-byte chunks
    for (int i = threadIdx.x; i < nvec; i += blockDim.x) {
        const float* gp = g + 4 * i;
        unsigned loff = (unsigned)(unsigned long long)(lds + 4 * i);
        asm volatile("global_load_async_to_lds_b128 %0, %1, off"
                     :: "v"(loff), "v"(gp) : "memory");
    }
    asm volatile("s_wait_asynccnt 0x0" ::: "memory");
    __syncthreads();
}

// ---------------------------------------------------------------------
// zero a float range (aggr + sums + sumsq each layer)
__global__ void zero_f32(float* __restrict__ p, size_t n) {
    size_t i = (size_t)blockIdx.x * blockDim.x + threadIdx.x;
    size_t stride = (size_t)gridDim.x * blockDim.x;
    for (; i < n; i += stride) p[i] = 0.0f;
}

// ---------------------------------------------------------------------
// h0[b,n,:] = x[b,n,:] @ inW + inb     (one 64-thread block per node)
__global__ void input_embed(const float* __restrict__ x,
                            const float* __restrict__ inW,
                            const float* __restrict__ inb,
                            float* __restrict__ h) {
    __shared__ float xs[INDIM];
    int node = blockIdx.x;          // 0 .. B*N-1
    int c = threadIdx.x;            // 0 .. 63
    if (c < INDIM) xs[c] = x[(size_t)node * INDIM + c];
    __syncthreads();
    float acc = inb[c];
#pragma unroll
    for (int k = 0; k < INDIM; ++k) acc = fmaf(xs[k], inW[k * EDIM + c], acc);
    h[(size_t)node * EDIM + c] = acc;
}

// ---------------------------------------------------------------------
// one wave32 per edge: aggr[dst, 0:64]   += h[src, :]
//                      aggr[dst, 64:128] += a*edgeW + edgeb
__global__ void scatter_edges(const float* __restrict__ h,
                              const long long* __restrict__ eidx,  // [B,2,E] i64
                              const float* __restrict__ eattr,     // [B,E]
                              const float* __restrict__ eW,        // [64] (layer slice)
                              const float* __restrict__ eb,        // [64]
                              float* __restrict__ aggr) {
    long long wid = ((long long)blockIdx.x * blockDim.x + threadIdx.x) >> 5;
    int lane = threadIdx.x & 31;
    if (wid >= (long long)BGR * NEDGE) return;
    int b = (int)(wid / NEDGE);
    int e = (int)(wid % NEDGE);
    const long long* ei = eidx + (size_t)b * 2 * NEDGE;
    int src = (int)ei[e];
    int dst = (int)ei[NEDGE + e];
    float a = eattr[(size_t)b * NEDGE + e];
    const float* hsrc = h + ((size_t)b * NNODE + src) * EDIM;
    float* out = aggr + ((size_t)b * NNODE + dst) * H2;
#pragma unroll
    for (int i = 0; i < 2; ++i) {
        int c = lane + 32 * i;
        atomicAdd(&out[c], hsrc[c]);
        atomicAdd(&out[EDIM + c], fmaf(a, eW[c], eb[c]));
    }
}

// ---------------------------------------------------------------------
// t = aggr @ W1 + b1, computed IN-PLACE (T == A is safe: each wave reads
// only its own 16 rows fully into registers before writing them back).
// fp32 WMMA 16x16x4. Register-blocked: 8 independent accumulators so
// consecutive WMMAs have no RAW chain and DS loads pipeline under them.
__global__ void gemm1_wmma(const float* __restrict__ A,
                           const float* __restrict__ W1,   // [128,128] layer slice
                           const float* __restrict__ b1,   // [128]
                           float* __restrict__ T) {
    __shared__ float Ws[H2 * H2];                 // 64 KB of the 320 KB/WGP LDS
    async_stage(W1, Ws, H2 * H2);

    int wave = threadIdx.x >> 5;
    int lane = threadIdx.x & 31;
    int rb = blockIdx.x * (blockDim.x >> 5) + wave;       // 16-row block id
    if (rb >= (BGR * NNODE) / 16) return;
    int row0 = rb * 16;
    int m  = lane & 15;                                    // M (A) / N (B,D)
    int kh = lane >> 4;                                    // K-half select

    // A fragments for all K: af[s] = { A[m][4s+2kh], A[m][4s+2kh+1] }
    v2f af[32];
    const float* arow = A + (size_t)(row0 + m) * H2 + 2 * kh;
#pragma unroll
    for (int s = 0; s < 32; ++s) af[s] = *(const v2f*)(arow + 4 * s);

    // 8 output tiles held simultaneously (64 VGPRs)
    v8f acc[8];
#pragma unroll
    for (int nt = 0; nt < 8; ++nt) {
        float bias = b1[nt * 16 + m];
#pragma unroll
        for (int r = 0; r < 8; ++r) acc[nt][r] = bias;
    }

#pragma unroll
    for (int s = 0; s < 32; ++s) {
        int k = 4 * s + 2 * kh;
        const float* wrow = &Ws[k * H2 + m];
#pragma unroll
        for (int nt = 0; nt < 8; ++nt) {
            v2f bf;
            bf.x = wrow[nt * 16];
            bf.y = wrow[H2 + nt * 16];
            acc[nt] = __builtin_amdgcn_wmma_f32_16x16x4_f32(
                false, af[s], false, bf, (short)0, acc[nt], false, false);
        }
    }

#pragma unroll
    for (int nt = 0; nt < 8; ++nt) {
        float* trow = T + (size_t)row0 * H2 + nt * 16 + m;
#pragma unroll
        for (int r = 0; r < 8; ++r)
            trow[(size_t)(r + 8 * kh) * H2] = acc[nt][r];
    }
}

// ---------------------------------------------------------------------
// per-graph, per-channel sum / sumsq of t  (coalesced: 128 threads = 128 ch)
__global__ void bn_stats(const float* __restrict__ T,
                         float* __restrict__ sums,
                         float* __restrict__ sumsq) {
    const int CHUNKS = 160, NPB = NNODE / CHUNKS;   // 125 nodes/block
    int b = blockIdx.x / CHUNKS;
    int chunk = blockIdx.x % CHUNKS;
    int c = threadIdx.x;
    const float* base = T + ((size_t)b * NNODE + (size_t)chunk * NPB) * H2 + c;
    float s = 0.f, s2 = 0.f;
    for (int i = 0; i < NPB; ++i) {
        float v = base[(size_t)i * H2];
        s += v; s2 = fmaf(v, v, s2);
    }
    atomicAdd(&sums[b * H2 + c], s);
    atomicAdd(&sumsq[b * H2 + c], s2);
}

__global__ void bn_finalize(const float* __restrict__ sums,
                            const float* __restrict__ sumsq,
                            const float* __restrict__ gamma,
                            const float* __restrict__ beta,
                            float* __restrict__ scale,
                            float* __restrict__ shift) {
    int i = blockIdx.x * blockDim.x + threadIdx.x;   // 0 .. B*128-1
    if (i >= BGR * H2) return;
    int c = i % H2;
    float inv_n = 1.0f / (float)NNODE;
    float mean = sums[i] * inv_n;
    float var  = sumsq[i] * inv_n - mean * mean;
    float sc = gamma[c] * rsqrtf(var + BN_EPS);
    scale[i] = sc;
    shift[i] = beta[c] - mean * sc;
}

// ---------------------------------------------------------------------
// h_out = [maybe relu]( relu(bn(t)) @ W2 + b2 ); BN+ReLU fused into A-frag load
__global__ void gemm2_wmma(const float* __restrict__ T,
                           const float* __restrict__ W2,    // [128,64] layer slice
                           const float* __restrict__ b2,    // [64]
                           const float* __restrict__ scale, // [B,128]
                           const float* __restrict__ shift, // [B,128]
                           float* __restrict__ Hout,
                           int apply_relu) {
    __shared__ float Ws[H2 * EDIM];               // 32 KB
    async_stage(W2, Ws, H2 * EDIM);

    int wave = threadIdx.x >> 5;
    int lane = threadIdx.x & 31;
    int rb = blockIdx.x * (blockDim.x >> 5) + wave;
    if (rb >= (BGR * NNODE) / 16) return;
    int row0 = rb * 16;
    int b = row0 / NNODE;                 // 16 | NNODE, so no straddling
    int m  = lane & 15;
    int kh = lane >> 4;

    const float* sc  = scale + b * H2 + 2 * kh;
    const float* sh  = shift + b * H2 + 2 * kh;
    const float* trw = T + (size_t)(row0 + m) * H2 + 2 * kh;

    v2f af[32];
#pragma unroll
    for (int s = 0; s < 32; ++s) {
        v2f t  = *(const v2f*)(trw + 4 * s);
        v2f cs = *(const v2f*)(sc + 4 * s);
        v2f ch = *(const v2f*)(sh + 4 * s);
        v2f v;
        v.x = fmaxf(0.f, fmaf(t.x, cs.x, ch.x));
        v.y = fmaxf(0.f, fmaf(t.y, cs.y, ch.y));
        af[s] = v;
    }

    v8f acc[4];
#pragma unroll
    for (int nt = 0; nt < 4; ++nt) {
        float bias = b2[nt * 16 + m];
#pragma unroll
        for (int r = 0; r < 8; ++r) acc[nt][r] = bias;
    }

#pragma unroll
    for (int s = 0; s < 32; ++s) {
        int k = 4 * s + 2 * kh;
        const float* wrow = &Ws[k * EDIM + m];
#pragma unroll
        for (int nt = 0; nt < 4; ++nt) {
            v2f bf;
            bf.x = wrow[nt * 16];
            bf.y = wrow[EDIM + nt * 16];
            acc[nt] = __builtin_amdgcn_wmma_f32_16x16x4_f32(
                false, af[s], false, bf, (short)0, acc[nt], false, false);
        }
    }

#pragma unroll
    for (int nt = 0; nt < 4; ++nt) {
        float* orow = Hout + (size_t)row0 * EDIM + nt * 16 + m;
#pragma unroll
        for (int r = 0; r < 8; ++r) {
            float v = acc[nt][r];
            if (apply_relu) v = fmaxf(0.f, v);
            orow[(size_t)(r + 8 * kh) * EDIM] = v;
        }
    }
}

// ---------------------------------------------------------------------
extern "C" void kernel_launch(void* const* d_in, const int* in_sizes, int n_in,
                              void* d_out, int out_size, void* d_ws, size_t ws_size,
                              hipStream_t stream) {
    (void)in_sizes; (void)n_in; (void)out_size; (void)ws_size;

    const float*     x     = (const float*)d_in[0];
    const long long* eidx  = (const long long*)d_in[1];   // int64 per reference
    const float*     eattr = (const float*)d_in[2];
    const float*     inW   = (const float*)d_in[3];
    const float*     inb   = (const float*)d_in[4];
    const float*     edgeW = (const float*)d_in[5];       // [L,1,64]
    const float*     edgeb = (const float*)d_in[6];       // [L,64]
    const float*     W1    = (const float*)d_in[7];       // [L,128,128]
    const float*     b1    = (const float*)d_in[8];
    const float*     gamma = (const float*)d_in[9];
    const float*     beta  = (const float*)d_in[10];
    const float*     W2    = (const float*)d_in[11];      // [L,128,64]
    const float*     b2    = (const float*)d_in[12];
    float* out = (float*)d_out;

    // workspace carve-up (~61.5 MB):
    // [aggr/t: B*N*128][sums: B*128][sumsq: B*128][scale][shift][hbuf: B*N*64]
    float* aggr  = (float*)d_ws;
    float* sums  = aggr  + (size_t)BGR * NNODE * H2;
    float* sumsq = sums  + BGR * H2;
    float* scale = sumsq + BGR * H2;
    float* shift = scale + BGR * H2;
    float* hbuf  = shift + BGR * H2;

    const int nRB = (BGR * NNODE) / 16;        // 5000 row blocks
    const int gGemm = nRB / 8;                  // 8 waves (256 thr) per block
    const long long nWaves = (long long)BGR * NEDGE;
    const int gScat = (int)((nWaves + 7) / 8);  // 8 edges per 256-thr block

    input_embed<<<BGR * NNODE, 64, 0, stream>>>(x, inW, inb, hbuf);

    for (int l = 0; l < LNUM; ++l) {
        // zero aggr + sums + sumsq (contiguous)
        zero_f32<<<2048, 256, 0, stream>>>(aggr, (size_t)BGR * NNODE * H2 + 2u * BGR * H2);

        scatter_edges<<<gScat, 256, 0, stream>>>(hbuf, eidx, eattr,
                                                 edgeW + l * EDIM, edgeb + l * EDIM, aggr);

        gemm1_wmma<<<gGemm, 256, 0, stream>>>(aggr, W1 + (size_t)l * H2 * H2,
                                              b1 + l * H2, aggr);   // in-place

        bn_stats<<<BGR * 160, 128, 0, stream>>>(aggr, sums, sumsq);
        bn_finalize<<<1, 512, 0, stream>>>(sums, sumsq, gamma + l * H2, beta + l * H2,
                                           scale, shift);

        float* dst = (l == LNUM - 1) ? out : hbuf;
        gemm2_wmma<<<gGemm, 256, 0, stream>>>(aggr, W2 + (size_t)l * H2 * EDIM,
                                              b2 + l * EDIM, scale, shift, dst,
                                              (l < LNUM - 1) ? 1 : 0);
    }
}